// BNTM_Softmax_7421703487926
// MI455X (gfx1250) — compile-verified
//
#include <hip/hip_runtime.h>
#include <hip/hip_bf16.h>
#include <math.h>

#define HID 2048
#define OUT 4096
#define VOC 4096
#define MEMN 4096
#define MDIM 64
#define OPN 5

typedef __attribute__((ext_vector_type(2))) float v2f;
typedef __attribute__((ext_vector_type(8))) float v8f;

// ---------------------------------------------------------------------------
// WMMA f32 16x16x4 GEMV core: one wave computes 16 output rows of y = W * x.
// A-tile layout (ISA 7.12.2, 32-bit A 16x4): lanes 0-15 hold K=kbase+{0,1} in
// the two A VGPRs, lanes 16-31 hold K=kbase+{2,3}.  So each lane issues one
// 8-byte load from its row of W at column (kbase + kofs).  B is the x-chunk
// replicated across all 16 columns (B[k][n] = x[kbase+k] for every n), so all
// 16 columns of D carry identical partial dot products.
// ---------------------------------------------------------------------------
__device__ inline v8f wmma_gemv_acc(const float* __restrict__ w_lane,  // W + row*ldk + kofs
                                    const float* __restrict__ xv,     // vector
                                    int K, int kofs, v8f acc) {
  for (int k = 0; k < K; k += 4) {
    v2f a = *(const v2f*)(w_lane + k);       // W[row][k+kofs .. k+kofs+1]
    v2f b = *(const v2f*)(xv + k + kofs);    // x[k+kofs .. k+kofs+1] (bcast over N)
    acc = __builtin_amdgcn_wmma_f32_16x16x4_f32(
        /*neg_a=*/false, a, /*neg_b=*/false, b,
        /*c_mod=*/(short)0, acc, /*reuse_a=*/false, /*reuse_b=*/false);
  }
  return acc;
}

// ---------------------------------------------------------------------------
// K1: hbar[i] = dot(Wm_w[i,0:64], memory[0,0:64]) + Wm_b[i] + hidden0[i]
// ---------------------------------------------------------------------------
__global__ __launch_bounds__(256) void hbar_kernel(
    const float* __restrict__ mem0, const float* __restrict__ Wm,
    const float* __restrict__ bm, const float* __restrict__ h0,
    float* __restrict__ hbar) {
  int i = blockIdx.x * 256 + threadIdx.x;  // i < HID
  const float* w = Wm + (size_t)i * MDIM;
  float s = bm[i] + h0[i];
#pragma unroll 8
  for (int k = 0; k < MDIM; ++k) s += w[k] * mem0[k];
  hbar[i] = s;
}

// ---------------------------------------------------------------------------
// K2: h = tanh(W_ih @ x + b_ih + W_hh @ hbar + b_hh), 16 rows per wave.
// grid = HID / (8 waves * 16 rows) = 16 blocks of 256 threads.
// ---------------------------------------------------------------------------
__global__ __launch_bounds__(256) void rnn_cell_kernel(
    const float* __restrict__ x, const float* __restrict__ hbar,
    const float* __restrict__ Wih, const float* __restrict__ bih,
    const float* __restrict__ Whh, const float* __restrict__ bhh,
    float* __restrict__ h, float* __restrict__ ht_out) {
  const int lane = threadIdx.x & 31;
  const int wave = threadIdx.x >> 5;
  const int m0   = (blockIdx.x * 8 + wave) * 16;
  const int half = lane >> 4;            // 0: K pair {0,1}; 1: K pair {2,3}
  const int kofs = half * 2;
  const int row  = m0 + (lane & 15);

  v8f acc = {};
  acc = wmma_gemv_acc(Wih + (size_t)row * VOC + kofs, x,    VOC, kofs, acc);
  acc = wmma_gemv_acc(Whh + (size_t)row * HID + kofs, hbar, HID, kofs, acc);

  // D layout: VGPR r holds M=r (lanes 0-15) / M=r+8 (lanes 16-31); all N equal.
  if ((lane & 15) == 0) {
    int base = m0 + half * 8;
#pragma unroll
    for (int r = 0; r < 8; ++r) {
      float v = acc[r] + bih[base + r] + bhh[base + r];
      v = tanhf(v);
      h[base + r]      = v;
      ht_out[base + r] = v;   // ht output == h
    }
  }
}

// ---------------------------------------------------------------------------
// K3: output = sigmoid(Wy @ h + Wy_b), 16 rows per wave.  grid = 32 blocks.
// ---------------------------------------------------------------------------
__global__ __launch_bounds__(256) void out_head_kernel(
    const float* __restrict__ h, const float* __restrict__ Wy,
    const float* __restrict__ by, float* __restrict__ out_y) {
  const int lane = threadIdx.x & 31;
  const int wave = threadIdx.x >> 5;
  const int m0   = (blockIdx.x * 8 + wave) * 16;
  const int half = lane >> 4;
  const int kofs = half * 2;
  const int row  = m0 + (lane & 15);

  v8f acc = {};
  acc = wmma_gemv_acc(Wy + (size_t)row * HID + kofs, h, HID, kofs, acc);

  if ((lane & 15) == 0) {
    int base = m0 + half * 8;
#pragma unroll
    for (int r = 0; r < 8; ++r) {
      float v = acc[r] + by[base + r];
      out_y[base + r] = 1.0f / (1.0f + expf(-v));
    }
  }
}

// ---------------------------------------------------------------------------
// K4: a = softmax(Wa @ h + Wa_b) (5 logits), new_elt = sigmoid(Wn @ h + Wn_b)
// (64 values).  Single block (32 waves) so the softmax sees all logits.
// ---------------------------------------------------------------------------
__global__ __launch_bounds__(1024) void head_small_kernel(
    const float* __restrict__ h,
    const float* __restrict__ Wa, const float* __restrict__ ba,
    const float* __restrict__ Wn, const float* __restrict__ bn,
    float* __restrict__ a_out, float* __restrict__ ne_out) {
  __shared__ float logits[OPN];
  const int wave = threadIdx.x >> 5;
  const int lane = threadIdx.x & 31;

  for (int o = wave; o < OPN + MDIM; o += 32) {
    const float* wrow;
    float bias;
    if (o < OPN) { wrow = Wa + (size_t)o * HID;         bias = ba[o]; }
    else         { wrow = Wn + (size_t)(o - OPN) * HID; bias = bn[o - OPN]; }
    float s = 0.0f;
    for (int k = lane; k < HID; k += 32) s += wrow[k] * h[k];
#pragma unroll
    for (int off = 16; off > 0; off >>= 1) s += __shfl_xor(s, off);
    if (lane == 0) {
      float v = s + bias;
      if (o < OPN) logits[o] = v;
      else         ne_out[o - OPN] = 1.0f / (1.0f + expf(-v));
    }
  }
  __syncthreads();
  if (threadIdx.x == 0) {
    float mx = logits[0];
#pragma unroll
    for (int j = 1; j < OPN; ++j) mx = fmaxf(mx, logits[j]);
    float e[OPN], den = 0.0f;
#pragma unroll
    for (int j = 0; j < OPN; ++j) { e[j] = expf(logits[j] - mx); den += e[j]; }
#pragma unroll
    for (int j = 0; j < OPN; ++j) a_out[j] = e[j] / den;
  }
}

// ---------------------------------------------------------------------------
// K5: memory update.  roll/zero-row algebra collapses to per-row coefficients:
//   mem[r] = (a0 + a4*(r!=MEM-1)) * M[(r+1)%MEM]
//          + (a1 + a3*(r!=0))     * M[(r-1)%MEM]
//          +  a2                  * M[r]
//          + (r==0 ? new_elt : 0)
// ---------------------------------------------------------------------------
__global__ __launch_bounds__(256) void mem_update_kernel(
    const float* __restrict__ M, const float* __restrict__ a,
    const float* __restrict__ ne, float* __restrict__ out_m) {
  int idx = blockIdx.x * 256 + threadIdx.x;  // < MEMN*MDIM
  int r = idx >> 6;
  int c = idx & 63;
  float a0 = a[0], a1 = a[1], a2 = a[2], a3 = a[3], a4 = a[4];
  int up = (r + 1) & (MEMN - 1);
  int dn = (r + MEMN - 1) & (MEMN - 1);
  float cu = a0 + ((r != MEMN - 1) ? a4 : 0.0f);
  float cd = a1 + ((r != 0) ? a3 : 0.0f);
  float v = cu * M[(size_t)up * MDIM + c] + cd * M[(size_t)dn * MDIM + c] + a2 * M[idx];
  if (r == 0) v += ne[c];
  out_m[idx] = v;
}

extern "C" void kernel_launch(void* const* d_in, const int* in_sizes, int n_in,
                              void* d_out, int out_size, void* d_ws, size_t ws_size,
                              hipStream_t stream) {
  const float* x    = (const float*)d_in[0];   // input  (VOC)
  const float* h0   = (const float*)d_in[1];   // hidden0 (HID)
  const float* Mmem = (const float*)d_in[2];   // memory (MEMN, MDIM)
  const float* Wih  = (const float*)d_in[3];
  const float* bih  = (const float*)d_in[4];
  const float* Whh  = (const float*)d_in[5];
  const float* bhh  = (const float*)d_in[6];
  const float* Wm   = (const float*)d_in[7];
  const float* bm   = (const float*)d_in[8];
  const float* Wy   = (const float*)d_in[9];
  const float* by   = (const float*)d_in[10];
  const float* Wn   = (const float*)d_in[11];
  const float* bn   = (const float*)d_in[12];
  const float* Wa   = (const float*)d_in[13];
  const float* ba   = (const float*)d_in[14];

  float* out   = (float*)d_out;
  float* out_y  = out;               // 4096
  float* out_ht = out + OUT;         // 2048
  float* out_m  = out + OUT + HID;   // 4096*64

  float* ws   = (float*)d_ws;
  float* hbar = ws;                  // 2048
  float* h    = ws + HID;            // 2048
  float* av   = ws + 2 * HID;        // 5 (padded to 8)
  float* ne   = ws + 2 * HID + 8;    // 64

  hbar_kernel      <<<HID / 256, 256, 0, stream>>>(Mmem, Wm, bm, h0, hbar);
  rnn_cell_kernel  <<<HID / 128, 256, 0, stream>>>(x, hbar, Wih, bih, Whh, bhh, h, out_ht);
  out_head_kernel  <<<OUT / 128, 256, 0, stream>>>(h, Wy, by, out_y);
  head_small_kernel<<<1, 1024, 0, stream>>>(h, Wa, ba, Wn, bn, av, ne);
  mem_update_kernel<<<(MEMN * MDIM) / 256, 256, 0, stream>>>(Mmem, av, ne, out_m);
}